// RandomFGLClassifier_30227979829341
// MI455X (gfx1250) — compile-verified
//
#include <hip/hip_runtime.h>

typedef __attribute__((ext_vector_type(2))) float v2f;
typedef __attribute__((ext_vector_type(4))) float v4f;
typedef __attribute__((ext_vector_type(8))) float v8f;

// ---------------------------------------------------------------------------
// Problem constants (from the reference)
// ---------------------------------------------------------------------------
#define NB      32          // batch
#define N0      200000
#define N1      1024
#define N2      256
#define N3      32
#define NCLS    10

__device__ __forceinline__ int lower_bound_i(const int* __restrict__ a, int n, int key) {
    int lo = 0, hi = n;
    while (lo < hi) {
        int mid = (lo + hi) >> 1;
        if (a[mid] < key) lo = mid + 1; else hi = mid;
    }
    return lo;
}

// ---------------------------------------------------------------------------
// Weight normalization: W[r][c] = g[r] * V[r][c] / ||V[r]||  (one block / row)
// ---------------------------------------------------------------------------
__global__ void wn_kernel(const float* __restrict__ V, const float* __restrict__ g,
                          float* __restrict__ W, int cols) {
    __shared__ float red[256];
    const int r = blockIdx.x;
    const float* v = V + (size_t)r * cols;
    float s = 0.f;
    for (int c = threadIdx.x; c < cols; c += 256) { float x = v[c]; s += x * x; }
    red[threadIdx.x] = s;
    __syncthreads();
    for (int off = 128; off > 0; off >>= 1) {
        if (threadIdx.x < (unsigned)off) red[threadIdx.x] += red[threadIdx.x + off];
        __syncthreads();
    }
    const float scale = g[r] / sqrtf(red[0]);
    for (int c = threadIdx.x; c < cols; c += 256)
        W[(size_t)r * cols + c] = v[c] * scale;
}

// ---------------------------------------------------------------------------
// Transpose x (32 x N0) -> xT (N0 x 32): each edge gather becomes one 128B line
// ---------------------------------------------------------------------------
__global__ void transpose_x(const float* __restrict__ x, float* __restrict__ xT) {
    __shared__ float tile[32][33];
    const int node0 = blockIdx.x * 32;
    const int tb = threadIdx.x >> 5;   // 0..7
    const int tl = threadIdx.x & 31;
#pragma unroll
    for (int i = 0; i < 4; i++) {
        const int b = tb + i * 8;
        tile[b][tl] = x[(size_t)b * N0 + node0 + tl];
    }
    __syncthreads();
#pragma unroll
    for (int i = 0; i < 4; i++) {
        const int nl = tb + i * 8;
        xT[(size_t)(node0 + nl) * 32 + tl] = tile[tl][nl];
    }
}

// ---------------------------------------------------------------------------
// Layer 0 (C_in = 1): one block (8 waves) per destination node; sorted dst ->
// binary-searched edge segment.  Each wave covers 4 edges per iteration:
//   lane = (sub-edge 0..3) x (batch-quad 0..7), per-lane b128 load of xT.
// Then z1[n][b][o] = pooled[b] * W0n[o] + b0[o]  (node-major 32x32 tile)
// ---------------------------------------------------------------------------
__global__ void fgl0_kernel(const float* __restrict__ xT,
                            const int* __restrict__ src, const int* __restrict__ dst, int E,
                            const float* __restrict__ W0n, const float* __restrict__ b0,
                            float* __restrict__ z1) {
    __shared__ v4f red[32][8];      // [w*4+sub][batch-quad]
    __shared__ float s[32];
    const int n   = blockIdx.x;
    const int lo  = lower_bound_i(dst, E, n);
    const int hi  = lower_bound_i(dst, E, n + 1);
    const int cnt = hi - lo;
    const int w    = threadIdx.x >> 5;
    const int lane = threadIdx.x & 31;
    const int sub  = lane >> 3;     // which of 4 edges this lane helps with
    const int q    = lane & 7;      // batch quad: batches 4q .. 4q+3

    v4f acc = {0.f, 0.f, 0.f, 0.f};
    for (int i = w * 4 + sub; i < cnt; i += 32) {
        const int sn = src[lo + i];
        acc += *(const v4f*)(xT + (size_t)sn * 32 + q * 4);
    }
    red[w * 4 + sub][q] = acc;
    __syncthreads();
    if (threadIdx.x < 8) {          // one thread per batch quad
        v4f t = {0.f, 0.f, 0.f, 0.f};
#pragma unroll
        for (int i = 0; i < 32; i++) t += red[i][threadIdx.x];
        *(v4f*)(s + threadIdx.x * 4) = t;
    }
    __syncthreads();
    for (int f = threadIdx.x; f < 32 * 32; f += 256) {
        const int b = f >> 5, o = f & 31;
        z1[(size_t)n * (32 * 32) + f] = s[b] * W0n[o] + b0[o];
    }
}

// ---------------------------------------------------------------------------
// Generic FGL layer with WMMA einsum.
//   zin : (n_in, 32, CIN) node-major tiles (32*CIN floats contiguous, 16B-aligned)
//   pooled(32xCIN) = sum of zin[src] over sorted edge segment (b128 per thread),
//   OUT(32xCOUT) = pooled * Wn^T (+bias) via V_WMMA_F32_16X16X4_F32.
//   OUT_NBC=true  -> zout[n][b][c]            (feeds next FGL layer)
//   OUT_NBC=false -> zout[b][c*N3 + n]        (flat layout for the FC layer)
// ---------------------------------------------------------------------------
template <int CIN, int COUT, bool OUT_NBC>
__global__ void fgl_wmma_kernel(const float* __restrict__ zin,
                                const int* __restrict__ src, const int* __restrict__ dst, int E,
                                const float* __restrict__ Wn, const float* __restrict__ bias,
                                float* __restrict__ zout) {
    constexpr int TILE = 32 * CIN;          // floats per node tile
    constexpr int NV   = TILE / (256 * 4);  // v4f per thread (1 or 2)
    __shared__ v4f Pv[TILE / 4];
    float* P = (float*)Pv;

    const int n  = blockIdx.x;
    const int lo = lower_bound_i(dst, E, n);
    const int hi = lower_bound_i(dst, E, n + 1);

    v4f acc[NV];
#pragma unroll
    for (int j = 0; j < NV; j++) acc[j] = (v4f){0.f, 0.f, 0.f, 0.f};

    // 4-deep edge unroll: up to 4*NV independent b128 loads in flight
    int e = lo;
    for (; e + 3 < hi; e += 4) {
        const v4f* p0 = (const v4f*)(zin + (size_t)src[e]     * TILE);
        const v4f* p1 = (const v4f*)(zin + (size_t)src[e + 1] * TILE);
        const v4f* p2 = (const v4f*)(zin + (size_t)src[e + 2] * TILE);
        const v4f* p3 = (const v4f*)(zin + (size_t)src[e + 3] * TILE);
#pragma unroll
        for (int j = 0; j < NV; j++) {
            acc[j] += p0[threadIdx.x + j * 256];
            acc[j] += p1[threadIdx.x + j * 256];
            acc[j] += p2[threadIdx.x + j * 256];
            acc[j] += p3[threadIdx.x + j * 256];
        }
    }
    for (; e < hi; e++) {
        const v4f* p0 = (const v4f*)(zin + (size_t)src[e] * TILE);
#pragma unroll
        for (int j = 0; j < NV; j++) acc[j] += p0[threadIdx.x + j * 256];
    }
#pragma unroll
    for (int j = 0; j < NV; j++) Pv[threadIdx.x + j * 256] = acc[j];  // ds_store_b128
    __syncthreads();

    // ---- WMMA einsum: OUT[m][o] = sum_k P[m][k] * Wn[o][k] ----------------
    const int w    = threadIdx.x >> 5;
    const int lane = threadIdx.x & 31;
    const int half = lane >> 4;           // lanes 0-15: K=0,1   lanes 16-31: K=2,3
    const int lm   = lane & 15;

    constexpr int NT    = COUT / 16;      // N tiles
    constexpr int NPAIR = 2 * NT;         // (M tiles = 2) * NT; multiple of 8

    for (int p = w; p < NPAIR; p += 8) {  // uniform trip count across waves
        const int mt = p & 1;
        const int nt = p >> 1;
        v8f c = {0.f, 0.f, 0.f, 0.f, 0.f, 0.f, 0.f, 0.f};
#pragma unroll
        for (int k0 = 0; k0 < CIN; k0 += 4) {
            const int kk = k0 + 2 * half;
            v2f a, b;
            a.x = P[(mt * 16 + lm) * CIN + kk];
            a.y = P[(mt * 16 + lm) * CIN + kk + 1];
            const float* wrow = Wn + (size_t)(nt * 16 + lm) * CIN;
            b.x = wrow[kk];
            b.y = wrow[kk + 1];
            // 8 args: (neg_a, A, neg_b, B, c_mod, C, reuse_a, reuse_b)
            c = __builtin_amdgcn_wmma_f32_16x16x4_f32(
                    false, a, false, b, (short)0, c, false, false);
        }
#pragma unroll
        for (int r = 0; r < 8; r++) {
            const int M = mt * 16 + r + 8 * half;   // batch index
            const int N = nt * 16 + lm;             // out-channel index
            const float val = c[r] + bias[N];
            if (OUT_NBC)
                zout[(size_t)n * (32 * COUT) + (size_t)M * COUT + N] = val;
            else
                zout[(size_t)M * (COUT * N3) + (size_t)N * N3 + n] = val;
        }
    }
}

// ---------------------------------------------------------------------------
// FC: out[b][k] = flat[b] . Wfcn[k] + bfc[k];  flat = z3 in (b, c*32+n) layout
// ---------------------------------------------------------------------------
__global__ void fc_kernel(const float* __restrict__ flat, const float* __restrict__ Wfcn,
                          const float* __restrict__ bfc, float* __restrict__ out) {
    __shared__ float red[256];
    const int b = blockIdx.x;
    const v4f* f = (const v4f*)(flat + (size_t)b * 4096);
    for (int k = 0; k < NCLS; k++) {
        const v4f* wr = (const v4f*)(Wfcn + (size_t)k * 4096);
        float s = 0.f;
        for (int j = threadIdx.x; j < 1024; j += 256) {
            const v4f a = f[j], w4 = wr[j];
            s += a.x * w4.x + a.y * w4.y + a.z * w4.z + a.w * w4.w;
        }
        red[threadIdx.x] = s;
        __syncthreads();
        for (int off = 128; off > 0; off >>= 1) {
            if (threadIdx.x < (unsigned)off) red[threadIdx.x] += red[threadIdx.x + off];
            __syncthreads();
        }
        if (threadIdx.x == 0) out[b * NCLS + k] = red[0] + bfc[k];
        __syncthreads();
    }
}

// ---------------------------------------------------------------------------
extern "C" void kernel_launch(void* const* d_in, const int* in_sizes, int n_in,
                              void* d_out, int out_size, void* d_ws, size_t ws_size,
                              hipStream_t stream) {
    const float* x    = (const float*)d_in[0];
    const int*   src0 = (const int*)  d_in[1];
    const int*   dst0 = (const int*)  d_in[2];
    const int*   src1 = (const int*)  d_in[3];
    const int*   dst1 = (const int*)  d_in[4];
    const int*   src2 = (const int*)  d_in[5];
    const int*   dst2 = (const int*)  d_in[6];
    const float* V0   = (const float*)d_in[7];
    const float* g0   = (const float*)d_in[8];
    const float* b0   = (const float*)d_in[9];
    const float* V1   = (const float*)d_in[10];
    const float* g1   = (const float*)d_in[11];
    const float* b1   = (const float*)d_in[12];
    const float* V2   = (const float*)d_in[13];
    const float* g2   = (const float*)d_in[14];
    const float* b2   = (const float*)d_in[15];
    const float* Vfc  = (const float*)d_in[16];
    const float* gfc  = (const float*)d_in[17];
    const float* bfc  = (const float*)d_in[18];

    const int E0 = in_sizes[1];   // 2048000
    const int E1 = in_sizes[3];   // 262144
    const int E2 = in_sizes[5];   // 8192

    // workspace carve-up (floats; all 16B-aligned)
    float* ws   = (float*)d_ws;
    float* xT   = ws;                            // N0*32      = 6,400,000
    float* z1   = xT   + (size_t)N0 * 32;        // N1*32*32   = 1,048,576
    float* z2   = z1   + (size_t)N1 * 32 * 32;   // N2*32*64   =   524,288
    float* z3   = z2   + (size_t)N2 * 32 * 64;   // 32*128*32  =   131,072
    float* W0n  = z3   + (size_t)32 * 128 * 32;  // 32
    float* W1n  = W0n  + 32;                     // 64*32
    float* W2n  = W1n  + 64 * 32;                // 128*64
    float* Wfcn = W2n  + 128 * 64;               // 10*4096

    // 1) normalize weights
    wn_kernel<<<32,  256, 0, stream>>>(V0,  g0,  W0n,  1);
    wn_kernel<<<64,  256, 0, stream>>>(V1,  g1,  W1n,  32);
    wn_kernel<<<128, 256, 0, stream>>>(V2,  g2,  W2n,  64);
    wn_kernel<<<10,  256, 0, stream>>>(Vfc, gfc, Wfcn, 4096);

    // 2) transpose x -> node-major for coalesced edge gathers
    transpose_x<<<N0 / 32, 256, 0, stream>>>(x, xT);

    // 3) layer 0 (C=1, scalar einsum), one block per dst node, b128 gathers
    fgl0_kernel<<<N1, 256, 0, stream>>>(xT, src0, dst0, E0, W0n, b0, z1);

    // 4) layer 1: 32x32 tiles -> 32x64, b128 gathers + WMMA f32
    fgl_wmma_kernel<32, 64, true><<<N2, 256, 0, stream>>>(z1, src1, dst1, E1, W1n, b1, z2);

    // 5) layer 2: 32x64 tiles -> 32x128, b128 gathers + WMMA f32, FC-flat layout
    fgl_wmma_kernel<64, 128, false><<<N3, 256, 0, stream>>>(z2, src2, dst2, E2, W2n, b2, z3);

    // 6) FC head
    fc_kernel<<<NB, 256, 0, stream>>>(z3, Wfcn, bfc, (float*)d_out);
}